// GNNLayer_35364760715592
// MI455X (gfx1250) — compile-verified
//
#include <hip/hip_runtime.h>
#include <math.h>

// GAT-style GNN layer for MI455X (gfx1250, wave32).
// Pipeline: [zero ws] -> WMMA GEMM h=x@lin_w -> per-node attention dots ->
// edge segment-softmax (atomic max / exp-sum / weighted scatter-add) ->
// WMMA GEMM out=(agg+gnn_bias)@ff_w+ff_b -> BatchNorm(train) + ReLU in d_out.
// Both GEMMs use V_WMMA_F32_16X16X4_F32 (full f32 precision, 16x16 tiles,
// K stepped by 4). M=50000, K in {64,128}, Ncols in {128,64}: all multiples
// of 16, so tiles cover exactly and EXEC is all-ones for every WMMA.

typedef __attribute__((ext_vector_type(2))) float v2f;
typedef __attribute__((ext_vector_type(8))) float v8f;

#define HEADS 2
#define C_PER_HEAD 64
#define HID 128           // HEADS * C_PER_HEAD
#define NEG_SLOPE 0.2f
#define BN_EPS 1e-5f

// ---- sortable-uint encoding for float atomicMax -----------------------------
__device__ __forceinline__ unsigned f2sort(float f) {
  unsigned u = __float_as_uint(f);
  return (u & 0x80000000u) ? ~u : (u | 0x80000000u);
}
__device__ __forceinline__ float sort2f(unsigned u) {
  return (u & 0x80000000u) ? __uint_as_float(u & 0x7FFFFFFFu)
                           : __uint_as_float(~u);
}

// ---- zero-init scratch ------------------------------------------------------
__global__ void k_zero(float* __restrict__ p, long n) {
  long stride = (long)gridDim.x * blockDim.x;
  for (long i = (long)blockIdx.x * blockDim.x + threadIdx.x; i < n; i += stride)
    p[i] = 0.0f;
}

// ---- WMMA f32 GEMM: C[M,Ncols] = (A[M,K] + abias?) @ B[K,Ncols] + cbias? ----
// One wave computes one 16x16 output tile, accumulating K in steps of 4 with
// V_WMMA_F32_16X16X4_F32. Bias handling is a compile-time template flag so the
// inner loop is a branch-free load->wmma stream.
// A-frag layout (16x4 f32): lanes 0-15 hold M=lane, VGPR0=K0,VGPR1=K1;
//                           lanes 16-31 hold M=lane-16, VGPR0=K2,VGPR1=K3.
// B-frag layout (4x16 f32): mirrored (lanes 0-15: N=lane, K0/K1; 16-31: K2/K3).
// C/D layout (16x16 f32, 8 VGPRs): VGPR r -> row r (lanes<16) / r+8 (lanes>=16),
//                                  col = lane&15.
template <bool ABIAS, bool CBIAS>
__global__ void k_gemm_wmma_f32(const float* __restrict__ A,
                                const float* __restrict__ B,
                                const float* __restrict__ abias,   // len K
                                const float* __restrict__ cbias,   // len Ncols
                                float* __restrict__ C,
                                int M, int K, int Ncols) {
  const int ntn   = Ncols >> 4;
  const int tiles = ((M + 15) >> 4) * ntn;
  const int wave  = threadIdx.x >> 5;
  const int lane  = threadIdx.x & 31;
  const int tile  = blockIdx.x * (blockDim.x >> 5) + wave;
  if (tile >= tiles) return;                    // uniform per wave
  const int mtile = tile / ntn;
  const int ntile = tile % ntn;
  const int half  = lane >> 4;                  // 0 or 1
  const int l     = lane & 15;
  const int arow  = mtile * 16 + l;
  const int bcol  = ntile * 16 + l;

  const float* __restrict__ arow_p = A + (long)arow * K + half * 2;
  const float* __restrict__ ab_p   = ABIAS ? (abias + half * 2) : nullptr;
  const float* __restrict__ bcol_p = B + (long)half * 2 * Ncols + bcol;

  v8f acc = {};
#pragma unroll 4
  for (int k = 0; k < K; k += 4) {
    float2 a2 = *(const float2*)(arow_p + k);       // 8B-aligned (k even)
    if (ABIAS) {
      const float2 b2 = *(const float2*)(ab_p + k);
      a2.x += b2.x;
      a2.y += b2.y;
    }
    v2f av = {a2.x, a2.y};
    const float* bp = bcol_p + (long)k * Ncols;
    v2f bv = {bp[0], bp[Ncols]};
    acc = __builtin_amdgcn_wmma_f32_16x16x4_f32(
        /*neg_a=*/false, av, /*neg_b=*/false, bv,
        /*c_mod=*/(short)0, acc, /*reuse_a=*/false, /*reuse_b=*/false);
  }
  const float cb = CBIAS ? cbias[bcol] : 0.0f;
#pragma unroll
  for (int r = 0; r < 8; ++r) {
    const int row = mtile * 16 + r + half * 8;
    C[(long)row * Ncols + bcol] = acc[r] + cb;
  }
}

// ---- per-node attention dot products ---------------------------------------
// alpha_i[n,h] = <h[n,h,:],att_i[h]> + <emb[n,:],att_em_i[h]>  (same for _j)
__global__ void k_node_alpha(const float* __restrict__ h,
                             const float* __restrict__ emb,
                             const float* __restrict__ att_i,
                             const float* __restrict__ att_j,
                             const float* __restrict__ att_em_i,
                             const float* __restrict__ att_em_j,
                             float* __restrict__ ai, float* __restrict__ aj,
                             int N) {
  const int idx = blockIdx.x * blockDim.x + threadIdx.x;
  if (idx >= N * HEADS) return;
  const int n  = idx >> 1;
  const int hd = idx & 1;
  const float* hp = h + (long)n * HID + hd * C_PER_HEAD;
  const float* ep = emb + (long)n * C_PER_HEAD;
  const float* wi = att_i + hd * C_PER_HEAD;
  const float* wj = att_j + hd * C_PER_HEAD;
  const float* wei = att_em_i + hd * C_PER_HEAD;
  const float* wej = att_em_j + hd * C_PER_HEAD;
  float si = 0.f, sj = 0.f;
#pragma unroll 8
  for (int c = 0; c < C_PER_HEAD; ++c) {
    const float hv = hp[c], ev = ep[c];
    si = fmaf(hv, wi[c], fmaf(ev, wei[c], si));
    sj = fmaf(hv, wj[c], fmaf(ev, wej[c], sj));
  }
  ai[idx] = si;
  aj[idx] = sj;
}

__device__ __forceinline__ void edge_sd(int e, int E, const int* __restrict__ src,
                                        const int* __restrict__ dst, int& s, int& d) {
  if (e < E) { s = src[e]; d = dst[e]; }
  else       { s = d = e - E; }                // appended self-loops
}

// ---- pass 1: segment max (atomicMax on sortable uint) -----------------------
__global__ void k_edge_max(const int* __restrict__ src, const int* __restrict__ dst,
                           const float* __restrict__ ai, const float* __restrict__ aj,
                           unsigned* __restrict__ amaxU, int E, int N) {
  const int e = blockIdx.x * blockDim.x + threadIdx.x;
  if (e >= E + N) return;
  int s, d; edge_sd(e, E, src, dst, s, d);
#pragma unroll
  for (int hd = 0; hd < HEADS; ++hd) {
    float a = ai[d * HEADS + hd] + aj[s * HEADS + hd];
    a = a > 0.f ? a : NEG_SLOPE * a;           // leaky_relu
    atomicMax(&amaxU[d * HEADS + hd], f2sort(a));
  }
}

// ---- pass 2: segment exp-sum ------------------------------------------------
__global__ void k_edge_sum(const int* __restrict__ src, const int* __restrict__ dst,
                           const float* __restrict__ ai, const float* __restrict__ aj,
                           const unsigned* __restrict__ amaxU,
                           float* __restrict__ denom, int E, int N) {
  const int e = blockIdx.x * blockDim.x + threadIdx.x;
  if (e >= E + N) return;
  int s, d; edge_sd(e, E, src, dst, s, d);
#pragma unroll
  for (int hd = 0; hd < HEADS; ++hd) {
    float a = ai[d * HEADS + hd] + aj[s * HEADS + hd];
    a = a > 0.f ? a : NEG_SLOPE * a;
    const float ex = expf(a - sort2f(amaxU[d * HEADS + hd]));
    atomicAdd(&denom[d * HEADS + hd], ex);
  }
}

// ---- pass 3: weighted scatter-add, one wave32 per edge ----------------------
// Lane L handles the 4 channels [L*4, L*4+3] of the 128-wide concat; head =
// L>>4. Gather h[src] (L2-resident: 25.6 MB), f32 atomic-add into agg[dst].
__global__ void k_edge_agg(const int* __restrict__ src, const int* __restrict__ dst,
                           const float* __restrict__ ai, const float* __restrict__ aj,
                           const unsigned* __restrict__ amaxU,
                           const float* __restrict__ denom,
                           const float* __restrict__ h,
                           float* __restrict__ agg, int E, int N) {
  const int gid  = blockIdx.x * blockDim.x + threadIdx.x;
  const int e    = gid >> 5;
  const int lane = gid & 31;
  if (e >= E + N) return;
  int s, d; edge_sd(e, E, src, dst, s, d);
  const int hd = lane >> 4;
  float a = ai[d * HEADS + hd] + aj[s * HEADS + hd];
  a = a > 0.f ? a : NEG_SLOPE * a;
  const float ex = expf(a - sort2f(amaxU[d * HEADS + hd]));
  const float w  = ex / (denom[d * HEADS + hd] + 1e-16f);
  const float4 hv = *(const float4*)(h + (long)s * HID + lane * 4);
  float* ap = agg + (long)d * HID + lane * 4;
  atomicAdd(ap + 0, hv.x * w);
  atomicAdd(ap + 1, hv.y * w);
  atomicAdd(ap + 2, hv.z * w);
  atomicAdd(ap + 3, hv.w * w);
}

// ---- BatchNorm stats: per-block partials, one atomic per column per block ---
__global__ void k_bn_stats(const float* __restrict__ out,
                           float* __restrict__ bnsum, float* __restrict__ bnsq,
                           int N) {
  __shared__ float ssum[256];
  __shared__ float ssq[256];
  const int col = threadIdx.x & 63;
  const int rl  = threadIdx.x >> 6;            // 0..3
  const int rend = (int)(blockIdx.x + 1) * 512 < N ? (int)(blockIdx.x + 1) * 512 : N;
  float s = 0.f, q = 0.f;
  for (int r = blockIdx.x * 512 + rl; r < rend; r += 4) {
    const float v = out[(long)r * 64 + col];
    s += v; q += v * v;
  }
  ssum[threadIdx.x] = s; ssq[threadIdx.x] = q;
  __syncthreads();
  if (rl == 0) {
    s = ssum[col] + ssum[col + 64] + ssum[col + 128] + ssum[col + 192];
    q = ssq[col]  + ssq[col + 64]  + ssq[col + 128]  + ssq[col + 192];
    atomicAdd(&bnsum[col], s);
    atomicAdd(&bnsq[col], q);
  }
}

// ---- BatchNorm apply (training stats, biased var) + ReLU, in-place ----------
__global__ void k_bn_apply(float* __restrict__ out,
                           const float* __restrict__ bnsum,
                           const float* __restrict__ bnsq,
                           const float* __restrict__ gamma,
                           const float* __restrict__ beta, int N) {
  const long i = (long)blockIdx.x * blockDim.x + threadIdx.x;
  if (i >= (long)N * 64) return;
  const int col = (int)(i & 63);
  const float invN = 1.0f / (float)N;
  const float mean = bnsum[col] * invN;
  const float var  = bnsq[col] * invN - mean * mean;
  float v = (out[i] - mean) * rsqrtf(var + BN_EPS) * gamma[col] + beta[col];
  out[i] = v > 0.f ? v : 0.f;
}

extern "C" void kernel_launch(void* const* d_in, const int* in_sizes, int n_in,
                              void* d_out, int out_size, void* d_ws, size_t ws_size,
                              hipStream_t stream) {
  const float* x        = (const float*)d_in[0];
  const int*   ei       = (const int*)  d_in[1];
  const float* emb      = (const float*)d_in[2];
  const float* lin_w    = (const float*)d_in[3];
  const float* att_i    = (const float*)d_in[4];
  const float* att_j    = (const float*)d_in[5];
  const float* att_em_i = (const float*)d_in[6];
  const float* att_em_j = (const float*)d_in[7];
  const float* gnn_bias = (const float*)d_in[8];
  const float* ff_w     = (const float*)d_in[9];
  const float* ff_b     = (const float*)d_in[10];
  const float* gamma    = (const float*)d_in[11];
  const float* beta     = (const float*)d_in[12];

  const int N = in_sizes[0] / 64;     // 50000
  const int E = in_sizes[1] / 2;      // 600000
  const int Et = E + N;               // with self-loops
  const int* srcp = ei;               // edge_index[0,:]
  const int* dstp = ei + E;           // edge_index[1,:]
  float* out = (float*)d_out;

  // Workspace layout (~52.8 MB)
  float* h     = (float*)d_ws;                    // N*128
  float* agg   = h   + (size_t)N * HID;           // N*128 (zeroed)
  float* ai    = agg + (size_t)N * HID;           // N*2
  float* aj    = ai  + (size_t)N * HEADS;         // N*2
  unsigned* amaxU = (unsigned*)(aj + (size_t)N * HEADS);   // N*2 (zeroed; 0 = -inf sortable)
  float* denom = (float*)(amaxU + (size_t)N * HEADS);      // N*2 (zeroed)
  float* bnsum = denom + (size_t)N * HEADS;       // 64 (zeroed)
  float* bnsq  = bnsum + 64;                      // 64 (zeroed)

  // 1) zero agg .. bnsq (ai/aj included harmlessly; fully overwritten later)
  const long zn = (long)N * (HID + 4 * HEADS) + 128;
  k_zero<<<2048, 256, 0, stream>>>(agg, zn);

  // 2) h = x @ lin_w   (M=N, K=64, Ncols=128) — WMMA f32
  {
    const int tiles  = ((N + 15) >> 4) * (HID >> 4);
    const int blocks = (tiles + 3) / 4;           // 4 waves/block
    k_gemm_wmma_f32<false, false><<<blocks, 128, 0, stream>>>(
        x, lin_w, nullptr, nullptr, h, N, 64, HID);
  }

  // 3) per-node attention dots
  k_node_alpha<<<(N * HEADS + 255) / 256, 256, 0, stream>>>(
      h, emb, att_i, att_j, att_em_i, att_em_j, ai, aj, N);

  // 4) segment softmax over dst
  k_edge_max<<<(Et + 255) / 256, 256, 0, stream>>>(srcp, dstp, ai, aj, amaxU, E, N);
  k_edge_sum<<<(Et + 255) / 256, 256, 0, stream>>>(srcp, dstp, ai, aj, amaxU, denom, E, N);

  // 5) weighted scatter-add: one wave32 per edge
  {
    const long threads = (long)Et * 32;
    k_edge_agg<<<(int)((threads + 255) / 256), 256, 0, stream>>>(
        srcp, dstp, ai, aj, amaxU, denom, h, agg, E, N);
  }

  // 6) out = (agg + gnn_bias) @ ff_w + ff_b   (M=N, K=128, Ncols=64) — WMMA f32
  {
    const int tiles  = ((N + 15) >> 4) * (64 >> 4);
    const int blocks = (tiles + 3) / 4;
    k_gemm_wmma_f32<true, true><<<blocks, 128, 0, stream>>>(
        agg, ff_w, gnn_bias, ff_b, out, N, HID, 64);
  }

  // 7) BatchNorm (training stats) + ReLU, in-place on d_out
  k_bn_stats<<<(N + 511) / 512, 256, 0, stream>>>(out, bnsum, bnsq, N);
  k_bn_apply<<<(int)(((long)N * 64 + 255) / 256), 256, 0, stream>>>(
      out, bnsum, bnsq, gamma, beta, N);
}